// GradTreeBlock_54322746360311
// MI455X (gfx1250) — compile-verified
//
#include <hip/hip_runtime.h>
#include <hip/hip_bf16.h>

// ---------------------------------------------------------------------------
// GradTree forward, exact reduction of the reference:
//   * entmax15 + straight-through hard one-hot  ==> per-(e,i) argmax over N
//   * softsign + round straight-through         ==> hard bit (x >= sv*)
//   * product over path * one-hot p             ==> binary-tree traversal,
//     out[b,e,:] = leaf_classes[e, leaf*(b,e), :]
// Final contraction via V_WMMA_F32_16X16X4_F32 (bit-exact: A is 0/1 one-hot,
// so each accumulate adds either 0.0 or the exact f32 leaf value).
// Traffic ~28MB -> ~1.2us at 23.3 TB/s; compute negligible.
// gfx1250 paths: v_wmma_f32_16x16x4_f32 + global_load_async_to_lds_b128
// (ASYNCcnt) + wave32 shuffle reductions.
// ---------------------------------------------------------------------------

typedef float v2f __attribute__((ext_vector_type(2)));
typedef float v8f __attribute__((ext_vector_type(8)));

constexpr int B_    = 1024;
constexpr int N_    = 128;
constexpr int E_    = 256;
constexpr int DEPTH = 6;
constexpr int I_    = 63;   // 2^6 - 1 internal nodes
constexpr int L_    = 64;   // 2^6 leaves
constexpr int C_    = 10;

// ---------------------------------------------------------------------------
// Kernel 1: per (e,i) row: n* = argmax_n split_index_array (first occurrence),
// sv* = split_values[e,i,n*].  One wave32 per row, shuffle butterfly reduce.
// ---------------------------------------------------------------------------
__global__ void __launch_bounds__(256) k_argmax(const float* __restrict__ sidx,
                                                const float* __restrict__ sval,
                                                int* __restrict__ nstar,
                                                float* __restrict__ svstar) {
  int wave = (blockIdx.x * blockDim.x + threadIdx.x) >> 5;
  int lane = threadIdx.x & 31;
  if (wave >= E_ * I_) return;
  const float* row = sidx + (size_t)wave * N_;

  float v = row[lane];
  int   idx = lane;
#pragma unroll
  for (int j = 1; j < N_ / 32; ++j) {
    int n = lane + 32 * j;
    float t = row[n];
    if (t > v) { v = t; idx = n; }          // strict > keeps first occurrence
  }
#pragma unroll
  for (int m = 16; m >= 1; m >>= 1) {       // wave32 butterfly, ties -> low idx
    float ov = __shfl_xor(v, m, 32);
    int   oi = __shfl_xor(idx, m, 32);
    if (ov > v || (ov == v && oi < idx)) { v = ov; idx = oi; }
  }
  if (lane == 0) {
    nstar[wave]  = idx;
    svstar[wave] = sval[(size_t)wave * N_ + idx];
  }
}

// ---------------------------------------------------------------------------
// Kernel 2: tree traversal.  One thread per (b,e); node tables for this e
// staged in LDS.  bit = (x[b,n*] >= sv*)  (equality -> bit 1, matching
// round-half-to-even of softsign(0) in the reference).  Heap: 2*node+1+bit.
// ---------------------------------------------------------------------------
__global__ void __launch_bounds__(256) k_traverse(const float* __restrict__ x,
                                                  const int* __restrict__ nstar,
                                                  const float* __restrict__ svstar,
                                                  int* __restrict__ leaf_out) {
  __shared__ int   s_n[I_];
  __shared__ float s_v[I_];
  int e = blockIdx.y;
  int t = threadIdx.x;
  if (t < I_) { s_n[t] = nstar[e * I_ + t]; s_v[t] = svstar[e * I_ + t]; }
  __syncthreads();

  int b = blockIdx.x * blockDim.x + t;
  const float* xr = x + (size_t)b * N_;
  int node = 0, leaf = 0;
#pragma unroll
  for (int d = 0; d < DEPTH; ++d) {
    int bit = (xr[s_n[node]] >= s_v[node]) ? 1 : 0;
    leaf = (leaf << 1) | bit;
    node = 2 * node + 1 + bit;
  }
  leaf_out[(size_t)e * B_ + b] = leaf;
}

// ---------------------------------------------------------------------------
// Kernel 3: out[b,e,c] = sum_l p[b,e,l] * leaf_classes[e,l,c] via
// V_WMMA_F32_16X16X4_F32.  One wave per (e, 16-row b-tile); all 8 waves of a
// block share the same e (64 waves per e, 8 waves per block).
//
// Staging: lc[e] (640 f32) is async-copied raw into LDS with
// global_load_async_to_lds_b128 (ASYNCcnt), then rearranged into a
// zero-padded pair tile s_b[pair][col][2] so the inner loop needs exactly one
// aligned ds_load_b64 per K-chunk -- no divergent global loads in the loop,
// EXEC stays all-ones through every WMMA.
//
// Register layouts (ISA 7.12.2, wave32):
//   A 16x4 f32 : M = lane%16, K = 2*(lane/16) + vgpr   (one-hot p, built in regs)
//   B 4x16 f32 : N = lane%16, K = 2*(lane/16) + vgpr   (from s_b)
//   D 16x16 f32: N = lane%16, M = vgpr + 8*(lane/16)
// ---------------------------------------------------------------------------
__global__ void __launch_bounds__(256) k_leaf_wmma(const float* __restrict__ lc,
                                                   const int* __restrict__ leaf_in,
                                                   float* __restrict__ out) {
  __shared__ __align__(16) float s_raw[L_ * C_];        // 640 f32: raw lc[e]
  __shared__ __align__(16) float s_b[L_ / 2][16][2];    // padded K-pair tile

  int tid   = threadIdx.x;
  int lane  = tid & 31;
  int wave  = blockIdx.x * 8 + (tid >> 5);
  int e     = wave >> 6;        // uniform across the block
  int btile = wave & 63;
  int half  = lane >> 4;        // 0: lanes 0-15, 1: lanes 16-31
  int col   = lane & 15;        // M for A rows, N (class col) for B/D

  // ---- async-stage lc[e] into LDS: 160 x b128 (threads 0..159 = waves 0..4)
  const float* src = lc + (size_t)e * (L_ * C_);
  if (tid < (L_ * C_) / 4) {
    uint32_t ldsoff = (uint32_t)(uintptr_t)(&s_raw[tid * 4]); // low dword of a
                                                              // generic LDS ptr
                                                              // is the LDS offset
    uint64_t gaddr  = (uint64_t)(uintptr_t)(src + tid * 4);
    asm volatile("global_load_async_to_lds_b128 %0, %1, off"
                 :: "v"(ldsoff), "v"(gaddr) : "memory");
  }
  asm volatile("s_wait_asynccnt 0x0" ::: "memory");
  __syncthreads();

  // ---- build zero-padded pair tile: s_b[p][c] = {lc[2p, c], lc[2p+1, c]}
#pragma unroll
  for (int j = 0; j < 2; ++j) {                // 512 entries / 256 threads
    int idx  = tid + 256 * j;
    int pair = idx >> 4;
    int c    = idx & 15;
    s_b[pair][c][0] = (c < C_) ? s_raw[(2 * pair)     * C_ + c] : 0.0f;
    s_b[pair][c][1] = (c < C_) ? s_raw[(2 * pair + 1) * C_ + c] : 0.0f;
  }
  __syncthreads();

  int myleaf = leaf_in[e * B_ + btile * 16 + col];     // leaf for row M=col

  v8f acc = {};
#pragma unroll
  for (int kk = 0; kk < L_ / 4; ++kk) {
    int k0 = 4 * kk + 2 * half;                        // this lane-half's K pair
    v2f a;
    a.x = (myleaf == k0)     ? 1.0f : 0.0f;
    a.y = (myleaf == k0 + 1) ? 1.0f : 0.0f;
    v2f bm = *(const v2f*)&s_b[2 * kk + half][col][0]; // one ds_load_b64
    acc = __builtin_amdgcn_wmma_f32_16x16x4_f32(
        /*neg_a=*/false, a, /*neg_b=*/false, bm,
        /*c_mod=*/(short)0, acc, /*reuse_a=*/false, /*reuse_b=*/false);
  }

  if (col < C_) {
#pragma unroll
    for (int v = 0; v < 8; ++v) {
      int M = v + 8 * half;
      int b = btile * 16 + M;
      out[((size_t)b * E_ + e) * C_ + col] = acc[v];
    }
  }
}

// ---------------------------------------------------------------------------
extern "C" void kernel_launch(void* const* d_in, const int* in_sizes, int n_in,
                              void* d_out, int out_size, void* d_ws, size_t ws_size,
                              hipStream_t stream) {
  const float* x    = (const float*)d_in[0];   // [B,N]
  const float* sval = (const float*)d_in[1];   // [E,I,N]
  const float* sidx = (const float*)d_in[2];   // [E,I,N]
  const float* lc   = (const float*)d_in[3];   // [E,L,C]
  float* out = (float*)d_out;                  // [B,E,C]

  // workspace: nstar[E*I] ints, svstar[E*I] floats, leaf[E*B] ints (~1.15 MB)
  char* ws = (char*)d_ws;
  int*   nstar  = (int*)ws;                    ws += sizeof(int)   * E_ * I_;
  float* svstar = (float*)ws;                  ws += sizeof(float) * E_ * I_;
  int*   leaf   = (int*)ws;

  dim3 blk(256);
  // 16128 rows, 8 waves/block -> 2016 blocks
  k_argmax<<<dim3((E_ * I_ + 7) / 8), blk, 0, stream>>>(sidx, sval, nstar, svstar);
  // B/256 x E blocks
  k_traverse<<<dim3(B_ / 256, E_), blk, 0, stream>>>(x, nstar, svstar, leaf);
  // E * (B/16) = 16384 waves, 8/block -> 2048 blocks (exact, EXEC always full)
  k_leaf_wmma<<<dim3(E_ * (B_ / 16) / 8), blk, 0, stream>>>(lc, leaf, out);
}